// SpikeTruncatedMixtureModel_24223615550235
// MI455X (gfx1250) — compile-verified
//
#include <hip/hip_runtime.h>
#include <hip/hip_bf16.h>
#include <math.h>

typedef __attribute__((ext_vector_type(2))) float v2f;
typedef __attribute__((ext_vector_type(8))) float v8f;

#define NSPIKES 16384
#define NUNITS  256
#define RANK    5
#define NCFULL  64
#define NCO     12
#define NCAND   10
#define DD      60          // RANK*NCO
#define OUTW    61          // 1 count + 60 xsum
#define LOG2PI  1.8378770664093453f
#define WAVES_PER_BLOCK 8

__global__ void stmm_zero_kernel(float* __restrict__ out, int nelem) {
    int i = blockIdx.x * blockDim.x + threadIdx.x;
    if (i < nelem) out[i] = 0.0f;
}

// One wave32 handles one spike. 8 waves (256 threads) per block.
__global__ __launch_bounds__(256)
void stmm_kernel(const float* __restrict__ features,      // (N,60)
                 const float* __restrict__ mu,            // (256,5,64)
                 const float* __restrict__ Coo_inv,       // (128,60,60) symmetric
                 const float* __restrict__ Coo_logdet,    // (128,)
                 const float* __restrict__ log_prop,      // (256,)
                 const float* __restrict__ noise_lp_p,    // (1,)
                 const int*   __restrict__ cands,         // (N,10)
                 const int*   __restrict__ nid,           // (N,)
                 const int*   __restrict__ obs_ix,        // (128,12)
                 float*       __restrict__ out)           // (256,61)
{
    __shared__ float sW  [WAVES_PER_BLOCK][64 * 16];  // W = Cinv * Xc^T, row-major 64x16
    __shared__ float sXf [WAVES_PER_BLOCK][64];       // xf padded to 64
    __shared__ float sMah[WAVES_PER_BLOCK][16];       // diag(G)
    __shared__ int   sObs[WAVES_PER_BLOCK][16];       // obs_ix[nid]

    const int tid  = threadIdx.x;
    const int wave = tid >> 5;
    const int lane = tid & 31;
    const int m    = lane & 15;     // matrix row index within fragment (lanes 0-15 / 16-31)
    const int g    = lane >> 4;     // half-wave selector (K sub-pair)
    const int n    = blockIdx.x * WAVES_PER_BLOCK + wave;

    const int nidn = nid[n];
    const float* Cin = Coo_inv + (long)nidn * (DD * DD);

    // Warm L2/WGP$ with this spike's Cinv (14400B = 113 cache lines of 128B)
    for (int p = lane; p < 113; p += 32)
        __builtin_prefetch((const char*)Cin + p * 128, 0, 1);

    // Stage xf (zero-padded to 64) and obs indices into LDS
    for (int d = lane; d < 64; d += 32)
        sXf[wave][d] = (d < DD) ? features[(long)n * DD + d] : 0.0f;
    if (lane < NCO) sObs[wave][lane] = obs_ix[nidn * NCO + lane];
    __syncthreads();

    // ---- Build Xc (16x64, zero padded) directly in f32 A-fragment layout ----
    // A 16x4 f32 frag: lanes 0-15 hold row m, K={0,1}; lanes 16-31 hold row m, K={2,3}.
    // Rows 0-9: xf - nu[c]; row 10: xf (noise); rows 11-15: 0. Cols 60-63: 0.
    const int cu = (m < NCAND) ? cands[(long)n * NCAND + m] : 0;
    const float* mu_c = mu + (long)cu * (RANK * NCFULL);

    auto xcval = [&](int kk) -> float {
        if (kk >= DD || m > NCAND) return 0.0f;
        float x = sXf[wave][kk];
        if (m < NCAND) {
            int r = kk / NCO;
            int j = kk - r * NCO;
            x -= mu_c[r * NCFULL + sObs[wave][j]];
        }
        return x;
    };

    v2f a[16];
    #pragma unroll
    for (int f = 0; f < 16; ++f) {
        int k = 4 * f + 2 * g;
        a[f].x = xcval(k);
        a[f].y = xcval(k + 1);
    }
    // NOTE: in the 32-bit layouts, this A-fragment of Xc is exactly the
    // B-fragment of Xc^T (B 4x16: lanes 0-15 -> K={0,1}, lanes 16-31 -> K={2,3},
    // N = lane&15), so a[] doubles as the Xc^T operand below.

    // ---- W = Cinv (64x64 padded) @ Xc^T (64x16):  4 tiles x 15 K-steps ----
    #pragma unroll
    for (int i = 0; i < 4; ++i) {
        v8f acc = {0.f, 0.f, 0.f, 0.f, 0.f, 0.f, 0.f, 0.f};
        const int  mrow = i * 16 + m;            // Cinv row for this lane's A frag
        const bool rv   = (mrow < DD);
        const float* crow = Cin + mrow * DD;
        #pragma unroll
        for (int e0 = 0; e0 < DD; e0 += 4) {
            int e = e0 + 2 * g;                  // e, e+1 <= 59 always
            v2f af;
            af.x = rv ? crow[e]     : 0.0f;
            af.y = rv ? crow[e + 1] : 0.0f;
            acc = __builtin_amdgcn_wmma_f32_16x16x4_f32(
                false, af, false, a[e0 >> 2], (short)0, acc, false, false);
        }
        // C-layout store: VGPR r -> row (16i + r + 8g), col m
        #pragma unroll
        for (int r = 0; r < 8; ++r)
            sW[wave][(i * 16 + r + 8 * g) * 16 + m] = acc[r];
    }
    __syncthreads();

    // ---- G = Xc (16x64) @ W (64x16): 15 K-steps; mahal = diag(G) ----
    v8f gacc = {0.f, 0.f, 0.f, 0.f, 0.f, 0.f, 0.f, 0.f};
    #pragma unroll
    for (int k0 = 0; k0 < DD; k0 += 4) {
        v2f bf;
        bf.x = sW[wave][(k0 + 2 * g)     * 16 + m];
        bf.y = sW[wave][(k0 + 2 * g + 1) * 16 + m];
        gacc = __builtin_amdgcn_wmma_f32_16x16x4_f32(
            false, a[k0 >> 2], false, bf, (short)0, gacc, false, false);
    }

    // Extract diagonal of G from C-layout: (c,c) for c<8 at lane c, VGPR c;
    // for c>=8 at lane c+16, VGPR c-8.
    #pragma unroll
    for (int r = 0; r < 8; ++r) {
        if (g == 0 && m == r)     sMah[wave][r]     = gacc[r];
        if (g == 1 && m == r + 8) sMah[wave][r + 8] = gacc[r];
    }
    __syncthreads();

    // ---- Softmax over 11 log-likelihoods (redundant per lane, 11 elems) ----
    const float base     = Coo_logdet[nidn] + (float)DD * LOG2PI;
    const float noise_lp = noise_lp_p[0];
    float ll[NCAND + 1];
    #pragma unroll
    for (int c = 0; c < NCAND; ++c)
        ll[c] = log_prop[cands[(long)n * NCAND + c]] - 0.5f * (base + sMah[wave][c]);
    ll[NCAND] = noise_lp - 0.5f * (base + sMah[wave][NCAND]);

    float mx = ll[0];
    #pragma unroll
    for (int c = 1; c <= NCAND; ++c) mx = fmaxf(mx, ll[c]);
    float q[NCAND + 1];
    float s = 0.0f;
    #pragma unroll
    for (int c = 0; c <= NCAND; ++c) { q[c] = __expf(ll[c] - mx); s += q[c]; }
    const float inv_s = 1.0f / s;

    // ---- Segment-sum: 10 cands x 61 entries = 610 atomics spread over lanes ----
    for (int t = lane; t < NCAND * OUTW; t += 32) {
        int c = t / OUTW;
        int d = t - c * OUTW;
        int u = cands[(long)n * NCAND + c];
        float qq = q[c] * inv_s;
        float val = (d == 0) ? qq : qq * sXf[wave][d - 1];
        atomicAdd(&out[u * OUTW + d], val);
    }
}

extern "C" void kernel_launch(void* const* d_in, const int* in_sizes, int n_in,
                              void* d_out, int out_size, void* d_ws, size_t ws_size,
                              hipStream_t stream) {
    const float* features   = (const float*)d_in[0];
    const float* mu         = (const float*)d_in[1];
    const float* Coo_inv    = (const float*)d_in[2];
    const float* Coo_logdet = (const float*)d_in[3];
    const float* log_prop   = (const float*)d_in[4];
    const float* noise_lp   = (const float*)d_in[5];
    const int*   cands      = (const int*)d_in[6];
    const int*   nid        = (const int*)d_in[7];
    const int*   obs_ix     = (const int*)d_in[8];
    float*       out        = (float*)d_out;

    stmm_zero_kernel<<<(out_size + 255) / 256, 256, 0, stream>>>(out, out_size);
    stmm_kernel<<<NSPIKES / WAVES_PER_BLOCK, 256, 0, stream>>>(
        features, mu, Coo_inv, Coo_logdet, log_prop, noise_lp,
        cands, nid, obs_ix, out);
}